// Prediction_27925877359067
// MI455X (gfx1250) — compile-verified
//
#include <hip/hip_runtime.h>
#include <cstdint>

// ---------------- problem constants (match reference) ----------------
#define BATCH      16
#define N1         27648      // 96*96*3
#define N2         6912       // 48*48*3
#define N3         1728       // 24*24*3
#define NTOT       (N1 + N2 + N3)       // 36288
#define NCLS       3
#define MTOT       (NTOT * NCLS)        // 108864
#define TOPK       500
#define SORTK      512
#define WORDS      (SORTK / 32)         // 16
#define NPROB      (BATCH * NCLS)       // 48
#define SCORE_TH   0.05f
#define NMS_TH     0.5f
#define NEG_BIG    3.0e38f

__device__ __forceinline__ float sigm(float x) { return 1.0f / (1.0f + expf(-x)); }

// ---------------------------------------------------------------------
// Kernel 1: decode all three scales -> ids / scores / bboxes, clear keep
// ---------------------------------------------------------------------
__global__ void yolo_decode_kernel(
    const float* __restrict__ o1, const float* __restrict__ o2, const float* __restrict__ o3,
    const float* __restrict__ a1, const float* __restrict__ a2, const float* __restrict__ a3,
    const float* __restrict__ f1, const float* __restrict__ f2, const float* __restrict__ f3,
    const float* __restrict__ s1, const float* __restrict__ s2, const float* __restrict__ s3,
    float* __restrict__ out_ids, float* __restrict__ out_scr, float* __restrict__ out_box,
    unsigned char* __restrict__ keep)
{
    int t = blockIdx.x * blockDim.x + threadIdx.x;
    if (t >= BATCH * NTOT) return;
    int b = t / NTOT;
    int n = t - b * NTOT;

    const float *o, *a, *f, *s;
    int ln, ns;
    if (n < N1)            { o = o1; a = a1; f = f1; s = s1; ln = n;            ns = N1; }
    else if (n < N1 + N2)  { o = o2; a = a2; f = f2; s = s2; ln = n - N1;       ns = N2; }
    else                   { o = o3; a = a3; f = f3; s = s3; ln = n - N1 - N2;  ns = N3; }

    const float4* orow = (const float4*)(o + ((size_t)b * ns + ln) * 8);
    float4 r0 = orow[0];            // tx ty tw th
    float4 r1 = orow[1];            // obj c0 c1 c2

    float ax = a[2 * ln + 0], ay = a[2 * ln + 1];
    float fx = f[2 * ln + 0], fy = f[2 * ln + 1];
    float sx = s[0],          sy = s[1];

    float cx = (sigm(r0.x) + fx) * sx;
    float cy = (sigm(r0.y) + fy) * sy;
    float hx = 0.5f * expf(r0.z) * ax;
    float hy = 0.5f * expf(r0.w) * ay;
    float4 bbox = make_float4(cx - hx, cy - hy, cx + hx, cy + hy);

    float obj = sigm(r1.x);
    float cls[NCLS] = { sigm(r1.y), sigm(r1.z), sigm(r1.w) };

    size_t base = (size_t)b * MTOT + (size_t)n * NCLS;
    float4* boxv = (float4*)out_box;
#pragma unroll
    for (int c = 0; c < NCLS; ++c) {
        float sc = obj * cls[c];
        bool pass = sc > SCORE_TH;
        size_t k = base + c;
        out_ids[k] = pass ? (float)c : -1.0f;
        out_scr[k] = pass ? sc       : -1.0f;
        boxv[k]    = bbox;
        keep[k]    = 0;
    }
}

// ---------------------------------------------------------------------
// Kernel 2: per-(batch,class) top-500 select + LDS bitonic sort
// ---------------------------------------------------------------------
#define NHIST 1024
#define HBASE 15616u   // (float bits >> 16) base for scores in (0.05, 1]

__global__ void topk_select_kernel(
    const float* __restrict__ scr_all,   // (B, M)
    const float* __restrict__ box_all,   // (B, M, 4)
    int*   __restrict__ tk_idx,          // (NPROB, SORTK)
    float* __restrict__ tk_scr,          // (NPROB, SORTK)
    float* __restrict__ tk_box)          // (NPROB, SORTK, 4)
{
    __shared__ unsigned hist[NHIST];
    __shared__ float    s_key[SORTK];
    __shared__ int      s_idx[SORTK];
    __shared__ int      sh_thr;
    __shared__ unsigned sh_cnt_hi, sh_need_mid, sh_ctr_hi, sh_ctr_mid;

    int bc = blockIdx.x;
    int b = bc / NCLS, c = bc - b * NCLS;
    int tid = threadIdx.x;

    hist[tid] = 0u; hist[tid + 512] = 0u;
    __syncthreads();

    // pass 1: histogram of positive (= passed-threshold class-c) scores
    for (int n = tid; n < NTOT; n += SORTK) {
        float sc = scr_all[(size_t)b * MTOT + (size_t)n * NCLS + c];
        if (sc > 0.0f) {
            unsigned bkt = (__float_as_uint(sc) >> 16) - HBASE;
            if (bkt > (NHIST - 1u)) bkt = NHIST - 1u;
            atomicAdd(&hist[bkt], 1u);
        }
    }
    __syncthreads();

    if (tid == 0) {
        unsigned acc = 0, cnt_hi = 0; int thr = 0; bool found = false;
        for (int k = NHIST - 1; k >= 0; --k) {
            unsigned h = hist[k];
            if (!found && acc + h >= TOPK) { thr = k; cnt_hi = acc; found = true; }
            acc += h;
        }
        if (!found) { thr = 0; cnt_hi = acc - hist[0]; }
        sh_thr      = thr;
        sh_cnt_hi   = cnt_hi;
        sh_need_mid = found ? (TOPK - cnt_hi) : hist[0];
        sh_ctr_hi   = 0u;
        sh_ctr_mid  = 0u;
    }
    s_key[tid] = -NEG_BIG;
    s_idx[tid] = -1;
    __syncthreads();

    // pass 2: compaction into <= TOPK slots
    int      thr      = sh_thr;
    unsigned cnt_hi   = sh_cnt_hi;
    unsigned need_mid = sh_need_mid;
    for (int n = tid; n < NTOT; n += SORTK) {
        float sc = scr_all[(size_t)b * MTOT + (size_t)n * NCLS + c];
        if (sc > 0.0f) {
            unsigned bkt = (__float_as_uint(sc) >> 16) - HBASE;
            if (bkt > (NHIST - 1u)) bkt = NHIST - 1u;
            int slot = -1;
            if ((int)bkt > thr) {
                slot = (int)atomicAdd(&sh_ctr_hi, 1u);
            } else if ((int)bkt == thr) {
                unsigned q = atomicAdd(&sh_ctr_mid, 1u);
                if (q < need_mid) slot = (int)(cnt_hi + q);
            }
            if (slot >= 0 && slot < SORTK) { s_key[slot] = sc; s_idx[slot] = n; }
        }
    }
    __syncthreads();

    // bitonic sort (descending, tie-break: smaller index first == stable argsort)
    for (int k = 2; k <= SORTK; k <<= 1) {
        for (int j = k >> 1; j > 0; j >>= 1) {
            int ixj = tid ^ j;
            if (ixj > tid) {
                float ka = s_key[tid], kb = s_key[ixj];
                int   ia = s_idx[tid], ib = s_idx[ixj];
                bool desc = ((tid & k) == 0);
                bool a_first = (ka > kb) || (ka == kb && ia < ib);
                bool doswap = desc ? !a_first : a_first;
                if (doswap) {
                    s_key[tid] = kb; s_key[ixj] = ka;
                    s_idx[tid] = ib; s_idx[ixj] = ia;
                }
            }
            __syncthreads();
        }
    }

    // write out sorted candidates + gathered boxes
    int ix = s_idx[tid];
    tk_idx[bc * SORTK + tid] = ix;
    tk_scr[bc * SORTK + tid] = s_key[tid];
    float4 bx = make_float4(0.f, 0.f, 0.f, 0.f);
    if (ix >= 0)
        bx = ((const float4*)box_all)[(size_t)b * MTOT + (size_t)ix * NCLS + c];
    ((float4*)tk_box)[bc * SORTK + tid] = bx;
}

// ---------------------------------------------------------------------
// Kernel 3: greedy NMS per (batch,class).
// Boxes are DMA'd into LDS with CDNA5 async global->LDS loads.
// ---------------------------------------------------------------------
__global__ void nms_kernel(
    const int*   __restrict__ tk_idx,
    const float* __restrict__ tk_box,
    unsigned char* __restrict__ keep)
{
    __shared__ float4        sbox[SORTK];           // 8 KB
    __shared__ float         sarea[SORTK];          // 2 KB
    __shared__ unsigned      srow[SORTK][WORDS];    // 32 KB: 512-bit row per candidate
    __shared__ unsigned      sremv[WORDS];
    __shared__ unsigned char skeep[SORTK];
    __shared__ unsigned char sval[SORTK];

    int bc = blockIdx.x;
    int b = bc / NCLS, c = bc - b * NCLS;
    int tid = threadIdx.x;

    int myidx = tk_idx[bc * SORTK + tid];
    sval[tid]  = (tid < TOPK && myidx >= 0) ? 1 : 0;
    skeep[tid] = 0;

    // --- CDNA5 async DMA: global -> LDS, 16B per candidate, bypasses VGPRs ---
    if (tid < TOPK) {
        unsigned ldsa = (unsigned)(uintptr_t)(&sbox[tid]);   // low 32 bits = LDS byte offset
        unsigned long long ga =
            (unsigned long long)(uintptr_t)(tk_box + (size_t)(bc * SORTK + tid) * 4);
        asm volatile("global_load_async_to_lds_b128 %0, %1, off"
                     :: "v"(ldsa), "v"(ga) : "memory");
    }
    asm volatile("s_wait_asynccnt 0" ::: "memory");
    __syncthreads();

    if (tid >= TOPK) sbox[tid] = make_float4(0.f, 0.f, 0.f, 0.f);
    {
        float4 bi = sbox[tid];
        sarea[tid] = fmaxf(bi.z - bi.x, 0.f) * fmaxf(bi.w - bi.y, 0.f);
    }
    __syncthreads();

    // suppression bit-rows: row[i] bit j  <=>  IoU(i, j) > NMS_TH
    {
        float4 bi = sbox[tid];
        float  ai = sarea[tid];
#pragma unroll 1
        for (int w = 0; w < WORDS; ++w) {
            unsigned bits = 0u;
#pragma unroll 8
            for (int jj = 0; jj < 32; ++jj) {
                int j = (w << 5) + jj;
                float4 bj = sbox[j];                 // wave-uniform j -> LDS broadcast
                float iw = fminf(bi.z, bj.z) - fmaxf(bi.x, bj.x);
                float ih = fminf(bi.w, bj.w) - fmaxf(bi.y, bj.y);
                float inter = fmaxf(iw, 0.f) * fmaxf(ih, 0.f);
                float uni = ai + sarea[j] - inter + 1e-12f;
                if (inter > NMS_TH * uni) bits |= (1u << jj);
            }
            srow[tid][w] = bits;
        }
    }
    __syncthreads();

    // greedy scan over the score-sorted order (sequential dependency, 1 lane)
    if (tid == 0) {
#pragma unroll
        for (int w = 0; w < WORDS; ++w) sremv[w] = 0u;
        for (int i = 0; i < TOPK; ++i) {
            bool suppressed = (sremv[i >> 5] >> (i & 31)) & 1u;
            if (sval[i] && !suppressed) {
                skeep[i] = 1;
#pragma unroll
                for (int w = 0; w < WORDS; ++w) sremv[w] |= srow[i][w];
            }
        }
    }
    __syncthreads();

    if (tid < TOPK && skeep[tid] && myidx >= 0)
        keep[(size_t)b * MTOT + (size_t)myidx * NCLS + c] = 1;
}

// ---------------------------------------------------------------------
// Kernel 4: apply suppression:  sup = (id >= 0) & !keep  ->  id=score=-1
// ---------------------------------------------------------------------
__global__ void finalize_kernel(
    float* __restrict__ ids, float* __restrict__ scr,
    const unsigned char* __restrict__ keep)
{
    int t = blockIdx.x * blockDim.x + threadIdx.x;
    if (t >= BATCH * MTOT) return;
    float id = ids[t];
    if (id >= 0.0f && keep[t] == 0) {
        ids[t] = -1.0f;
        scr[t] = -1.0f;
    }
}

// ---------------------------------------------------------------------
extern "C" void kernel_launch(void* const* d_in, const int* in_sizes, int n_in,
                              void* d_out, int out_size, void* d_ws, size_t ws_size,
                              hipStream_t stream)
{
    const float* o1 = (const float*)d_in[0];
    const float* o2 = (const float*)d_in[1];
    const float* o3 = (const float*)d_in[2];
    const float* a1 = (const float*)d_in[3];
    const float* a2 = (const float*)d_in[4];
    const float* a3 = (const float*)d_in[5];
    const float* f1 = (const float*)d_in[6];
    const float* f2 = (const float*)d_in[7];
    const float* f3 = (const float*)d_in[8];
    const float* s1 = (const float*)d_in[9];
    const float* s2 = (const float*)d_in[10];
    const float* s3 = (const float*)d_in[11];

    const size_t BM = (size_t)BATCH * MTOT;
    float* out_ids = (float*)d_out;                 // (B, M, 1)
    float* out_scr = out_ids + BM;                  // (B, M, 1)
    float* out_box = out_ids + 2 * BM;              // (B, M, 4)

    // workspace layout (all offsets 256B aligned)
    unsigned char* keep = (unsigned char*)d_ws;     // BM bytes
    size_t off = (BM + 255) & ~(size_t)255;
    int*   tk_idx = (int*)  ((char*)d_ws + off); off += (size_t)NPROB * SORTK * 4;
    float* tk_scr = (float*)((char*)d_ws + off); off += (size_t)NPROB * SORTK * 4;
    float* tk_box = (float*)((char*)d_ws + off); off += (size_t)NPROB * SORTK * 16;

    int nthreads = BATCH * NTOT;
    yolo_decode_kernel<<<(nthreads + 255) / 256, 256, 0, stream>>>(
        o1, o2, o3, a1, a2, a3, f1, f2, f3, s1, s2, s3,
        out_ids, out_scr, out_box, keep);

    topk_select_kernel<<<NPROB, SORTK, 0, stream>>>(
        out_scr, out_box, tk_idx, tk_scr, tk_box);

    nms_kernel<<<NPROB, SORTK, 0, stream>>>(tk_idx, tk_box, keep);

    finalize_kernel<<<((int)BM + 255) / 256, 256, 0, stream>>>(out_ids, out_scr, keep);
}